// RoiPoolingConv_49143015801174
// MI455X (gfx1250) — compile-verified
//
#include <hip/hip_runtime.h>
#include <hip/hip_bf16.h>
#include <stdint.h>

// ROI max-pool for MI455X (gfx1250).
// img: (1,200,200,512) f32 NHWC; rois: (1,128,4) f32 (x,y,w,h); out: (1,128,7,7,512) f32.
// Bandwidth-bound gather+max (~350 MB traffic ~= 15us @ 23.3 TB/s).
// CDNA5 path: per-wave async DMA pipeline global->LDS (ASYNCcnt), barrier-free.

#define POOL      7
#define IMG_H     200
#define IMG_W     200
#define IMG_C     512
#define NROIS     128
#define DEPTH     8                 // async pipeline depth (ring slots per wave)
#define NTHREADS  128               // 4 x wave32
#define NWAVES    4
#define SLOT_BYTES 512              // 32 lanes x 16B: one wave's channel slice of one pixel
#define PIX_BYTES  (IMG_C * 4)      // 2048 B per pixel (NHWC, C contiguous)
#define WAVE_LDS_BYTES (DEPTH * SLOT_BYTES)       // 4 KB per wave
#define TOTAL_LDS_BYTES (NWAVES * WAVE_LDS_BYTES) // 16 KB per block

typedef float v4f __attribute__((ext_vector_type(4)));

__global__ __launch_bounds__(NTHREADS) void roi_maxpool_async(
    const float* __restrict__ img,
    const float* __restrict__ rois,
    float* __restrict__ out)
{
    __shared__ float stage[TOTAL_LDS_BYTES / 4];

    const int tid  = (int)threadIdx.x;
    const int lane = tid & 31;
    const int wv   = tid >> 5;

    // Keep the LDS block allocated/alive; drain the DS write before any async
    // engine writes can land in the same bytes.
    stage[tid] = 0.0f;
    asm volatile("s_wait_dscnt 0x0" ::: "memory");

    const int b   = (int)blockIdx.x;          // b = r*49 + jy*7 + ix
    const int r   = b / (POOL * POOL);
    const int bin = b - r * (POOL * POOL);
    const int jy  = bin / POOL;
    const int ix  = bin - jy * POOL;

    const float rx = rois[r * 4 + 0];
    const float ry = rois[r * 4 + 1];
    const float rw = rois[r * 4 + 2];
    const float rh = rois[r * 4 + 3];

    // Match reference rounding exactly: step = w/P (one div), then k*step (mul),
    // then x+... (add), truncate toward zero. No FMA contraction.
    const float sx = __fdiv_rn(rw, (float)POOL);
    const float sy = __fdiv_rn(rh, (float)POOL);
    const int x1 = (int)__fadd_rn(rx, __fmul_rn((float)ix,       sx));
    const int x2 = (int)__fadd_rn(rx, __fmul_rn((float)(ix + 1), sx));
    const int y1 = (int)__fadd_rn(ry, __fmul_rn((float)jy,       sy));
    const int y2 = (int)__fadd_rn(ry, __fmul_rn((float)(jy + 1), sy));

    const int binw = x2 - x1;
    const int binh = y2 - y1;
    const int npix = (binw > 0 && binh > 0) ? (binw * binh) : 0;

    // This lane's byte offset inside a pixel's 2KB channel vector.
    const uint32_t chan_byte     = (uint32_t)(wv * SLOT_BYTES + lane * 16);
    // This lane's base byte offset inside its wave's LDS ring.
    const uint32_t lds_lane_base = (uint32_t)(wv * WAVE_LDS_BYTES + lane * 16);
    const uint64_t base          = (uint64_t)(uintptr_t)img;

    // Incremental issue state: raster order over the bin rectangle.
    // voff walks +2048 per pixel; at a row end it jumps to the next row start.
    uint32_t voff     = (uint32_t)((y1 * IMG_W + x1) * PIX_BYTES) + chan_byte;
    const uint32_t row_skip = (uint32_t)((IMG_W - binw) * PIX_BYTES);
    int      iss_col  = 0;   // column within the bin for the next issue
    uint32_t iss_slot = 0;   // ring slot for the next issue

    auto issue_next = [&]() {
        const uint32_t ldsa = lds_lane_base + iss_slot * SLOT_BYTES;
        // Async DMA: 16B per lane, 512B per wave per pixel, tracked by ASYNCcnt.
        asm volatile("global_load_async_to_lds_b128 %0, %1, %2"
                     :: "v"(ldsa), "v"(voff), "s"(base)
                     : "memory");
        voff += PIX_BYTES;
        if (++iss_col == binw) { iss_col = 0; voff += row_skip; }
        iss_slot = (iss_slot + 1) & (DEPTH - 1);
    };

    v4f acc = { -__builtin_inff(), -__builtin_inff(),
                -__builtin_inff(), -__builtin_inff() };
    uint32_t con_slot = 0;   // ring slot for the next consume

    auto consume_next = [&]() {
        const uint32_t ldsa = lds_lane_base + con_slot * SLOT_BYTES;
        con_slot = (con_slot + 1) & (DEPTH - 1);
        v4f v;
        asm volatile("ds_load_b128 %0, %1\n\t"
                     "s_wait_dscnt 0x0"
                     : "=v"(v) : "v"(ldsa) : "memory");
        acc.x = __builtin_fmaxf(acc.x, v.x);
        acc.y = __builtin_fmaxf(acc.y, v.y);
        acc.z = __builtin_fmaxf(acc.z, v.z);
        acc.w = __builtin_fmaxf(acc.w, v.w);
    };

    // Fill the pipeline.
    const int head = npix < DEPTH ? npix : DEPTH;
    for (int i = 0; i < head; ++i) issue_next();

    // Steady state: exactly DEPTH requests in flight -> wait asynccnt<=DEPTH-1
    // guarantees the oldest outstanding pixel has landed in LDS.
    int p = 0;
    for (; p + DEPTH < npix; ++p) {
        asm volatile("s_wait_asynccnt 0x7" ::: "memory");
        consume_next();
        issue_next();
    }
    // Tail: drain everything, then consume the remaining slots.
    asm volatile("s_wait_asynccnt 0x0" ::: "memory");
    for (; p < npix; ++p) consume_next();

    // out[r][jy][ix][c], c = 4*tid .. 4*tid+3
    v4f* o = (v4f*)(out + (size_t)b * IMG_C + (size_t)tid * 4);
    *o = acc;
}

extern "C" void kernel_launch(void* const* d_in, const int* in_sizes, int n_in,
                              void* d_out, int out_size, void* d_ws, size_t ws_size,
                              hipStream_t stream) {
    (void)in_sizes; (void)n_in; (void)out_size; (void)d_ws; (void)ws_size;
    const float* img  = (const float*)d_in[0];
    const float* rois = (const float*)d_in[1];
    float* out        = (float*)d_out;
    // d_in[2] is pool_size == 7, baked in as POOL.
    dim3 grid(NROIS * POOL * POOL);   // 6272 blocks, one per output bin
    dim3 block(NTHREADS);
    roi_maxpool_async<<<grid, block, 0, stream>>>(img, rois, out);
}